// DREGN_CF_73821897883701
// MI455X (gfx1250) — compile-verified
//
#include <hip/hip_runtime.h>

typedef float v2f __attribute__((ext_vector_type(2)));
typedef float v8f __attribute__((ext_vector_type(8)));

#define D 64          // feature dim (fixed by reference)
#define NLAYERS 3

// ---------- x = concat(user_emb, item_emb); acc = x ----------
__global__ void k_init(const float* __restrict__ ue, const float* __restrict__ ie,
                       float* __restrict__ x, float* __restrict__ acc,
                       long long usz, long long total) {
  long long i = (long long)blockIdx.x * blockDim.x + threadIdx.x;
  if (i >= total) return;
  float v = (i < usz) ? ue[i] : ie[i - usz];
  x[i] = v;
  acc[i] = v;
}

// ---------- zero a buffer (float4 granularity) ----------
__global__ void k_zero4(float4* __restrict__ p, long long n4) {
  long long i = (long long)blockIdx.x * blockDim.x + threadIdx.x;
  if (i >= n4) return;
  p[i] = make_float4(0.f, 0.f, 0.f, 0.f);
}

// ---------- SpMM scatter: x_next[dst] += w * x_cur[src] ----------
// 16 threads per edge, each thread handles 4 consecutive floats (float4 load,
// 4 x global_atomic_add_f32). Atomics land in L2 (both buffers < 192MB L2).
__global__ void k_scatter(const float* __restrict__ xin, float* __restrict__ xout,
                          const float* __restrict__ eval, const int* __restrict__ esrc,
                          const int* __restrict__ edst, long long nnz) {
  long long tid = (long long)blockIdx.x * blockDim.x + threadIdx.x;
  long long e = tid >> 4;
  int q = (int)(tid & 15);
  if (e >= nnz) return;
  int src = esrc[e];
  int dst = edst[e];
  float w = eval[e];
  const float4 v = *(const float4*)(xin + (long long)src * D + q * 4);
  float* o = xout + (long long)dst * D + q * 4;
  atomicAdd(o + 0, w * v.x);
  atomicAdd(o + 1, w * v.y);
  atomicAdd(o + 2, w * v.z);
  atomicAdd(o + 3, w * v.w);
}

// ---------- acc += x_next ----------
__global__ void k_accadd(float4* __restrict__ acc, const float4* __restrict__ xn,
                         long long n4) {
  long long i = (long long)blockIdx.x * blockDim.x + threadIdx.x;
  if (i >= n4) return;
  float4 a = acc[i];
  float4 b = xn[i];
  a.x += b.x; a.y += b.y; a.z += b.z; a.w += b.w;
  acc[i] = a;
}

// ---------- gather users rows, fold 1/(L+1)^2 = 1/16 scale into A ----------
__global__ void k_gather(const float* __restrict__ acc, const int* __restrict__ users,
                         float* __restrict__ ag, long long total) {
  long long i = (long long)blockIdx.x * blockDim.x + threadIdx.x;
  if (i >= total) return;
  int b = (int)(i >> 6);          // row in batch
  int k = (int)(i & (D - 1));
  ag[i] = acc[(long long)users[b] * D + k] * 0.0625f;
}

// ---------- C(M,N) = A(M,64) * B(N,64)^T via V_WMMA_F32_16X16X4_F32 ----------
// One wave32 computes a 16x16 C tile; K=64 -> 16 chained k=4 WMMAs.
// A frag (16x4 f32, 2 VGPRs): lane%16 = M row; lanes<16 hold K={0,1}, lanes>=16 K={2,3}.
// B frag (4x16 f32, 2 VGPRs): lane%16 = N col; same half-wave K split.
// C/D (8 VGPRs): lane%16 = N col; VGPR v = M row v (lanes<16) / v+8 (lanes>=16).
__global__ void k_gemm(const float* __restrict__ A, const float* __restrict__ B,
                       float* __restrict__ C, int M, int N) {
  const int lane  = threadIdx.x & 31;
  const int wave  = threadIdx.x >> 5;
  const int ntile = blockIdx.x * 8 + wave;
  const int mtile = blockIdx.y;
  if (ntile * 16 >= N) return;                 // wave-uniform: EXEC stays all-ones
  const int m0 = mtile * 16;
  const int n0 = ntile * 16;
  const int half  = lane >> 4;                 // 0: K lo pair, 1: K hi pair
  const int idx16 = lane & 15;
  const int mrow = (m0 + idx16 < M) ? (m0 + idx16) : (M - 1);   // clamped load
  const int ncol = (n0 + idx16 < N) ? (n0 + idx16) : (N - 1);
  const float* arow = A + (long long)mrow * D;
  const float* brow = B + (long long)ncol * D;  // B stored row-major (N,64): Bmat[k][n]=B[n][k]
  v8f c = {};
#pragma unroll
  for (int ks = 0; ks < 16; ++ks) {
    const int kb = ks * 4 + half * 2;
    const float2 av = *(const float2*)(arow + kb);
    const float2 bv = *(const float2*)(brow + kb);
    v2f a; a.x = av.x; a.y = av.y;
    v2f b; b.x = bv.x; b.y = bv.y;
    // 8 args: (neg_a, A, neg_b, B, c_mod, C, reuse_a, reuse_b)
    c = __builtin_amdgcn_wmma_f32_16x16x4_f32(false, a, false, b, (short)0, c,
                                              false, false);
  }
  // Store epilogue. Fast path (wave-uniform full tile): one base pointer,
  // 8 unguarded coalesced b32 stores at row stride N -- no per-element
  // exec-mask surgery or 64-bit muls in the store-bandwidth-bound kernel.
  if (m0 + 16 <= M && n0 + 16 <= N) {
    float* cp = C + (long long)(m0 + half * 8) * N + (n0 + idx16);
    const long long stride = N;
#pragma unroll
    for (int v = 0; v < 8; ++v) {
      cp[0] = c[v];
      cp += stride;
    }
  } else {
#pragma unroll
    for (int v = 0; v < 8; ++v) {
      const int m = m0 + v + half * 8;
      const int n = n0 + idx16;
      if (m < M && n < N) C[(long long)m * N + n] = c[v];
    }
  }
}

extern "C" void kernel_launch(void* const* d_in, const int* in_sizes, int n_in,
                              void* d_out, int out_size, void* d_ws, size_t ws_size,
                              hipStream_t stream) {
  const float* ue    = (const float*)d_in[0];   // (n_users, 64)
  const float* ie    = (const float*)d_in[1];   // (n_items, 64)
  const float* ev    = (const float*)d_in[2];   // (nnz,)
  const int*   es    = (const int*)d_in[3];     // (nnz,)
  const int*   ed    = (const int*)d_in[4];     // (nnz,)
  const int*   users = (const int*)d_in[5];     // (batch,)

  const long long usz   = in_sizes[0];          // n_users*64
  const long long isz   = in_sizes[1];          // n_items*64
  const long long nnz   = in_sizes[2];
  const int       batch = in_sizes[5];
  const int n_users = (int)(usz / D);
  const long long total = usz + isz;            // n_nodes*64 floats
  const int N = (int)(isz / D);                 // n_items
  const int M = batch;

  float* acc = (float*)d_ws;
  float* xc  = acc + total;
  float* xn  = xc + total;
  float* ag  = xn + total;                      // (batch, 64), pre-scaled

  const int T = 256;
  k_init<<<(unsigned)((total + T - 1) / T), T, 0, stream>>>(ue, ie, xc, acc, usz, total);

  const long long n4  = total / 4;
  const long long sth = nnz * 16;               // 16 threads/edge
  for (int l = 0; l < NLAYERS; ++l) {
    k_zero4<<<(unsigned)((n4 + T - 1) / T), T, 0, stream>>>((float4*)xn, n4);
    k_scatter<<<(unsigned)((sth + T - 1) / T), T, 0, stream>>>(xc, xn, ev, es, ed, nnz);
    k_accadd<<<(unsigned)((n4 + T - 1) / T), T, 0, stream>>>((float4*)acc,
                                                             (const float4*)xn, n4);
    float* t = xc; xc = xn; xn = t;             // host-side ping-pong (deterministic)
  }

  const long long gth = (long long)batch * D;
  k_gather<<<(unsigned)((gth + T - 1) / T), T, 0, stream>>>(acc, users, ag, gth);

  const float* items = acc + (long long)n_users * D;   // (n_items, 64)
  dim3 grid((unsigned)((N / 16 + 7) / 8), (unsigned)((M + 15) / 16));
  k_gemm<<<grid, 256, 0, stream>>>(ag, items, (float*)d_out, M, N);
}